// RPN_80616536146778
// MI455X (gfx1250) — compile-verified
//
// RPN proposal generation (decode + per-level top-k + sort + NMS + top-1000)
// for MI455X / gfx1250: wave32 NMS mask words, 128KB-LDS single-WGP bitonic
// sort, async global->LDS tile staging + global prefetch (gfx1250 paths).
#include <hip/hip_runtime.h>
#include <hip/hip_bf16.h>
#include <stdint.h>

#define IMG_WH    800.0f
#define BBOX_CLIPV 4.135166556742356f   /* log(1000/16) */
#define MIN_SIZE  0.001f
#define NMS_TH    0.7f
#define NEG_INF   (-__builtin_inff())

#define A_TOTAL 159882
#define KTOT    8382
#define KPAD    8384
#define NW      262        /* ceil(KTOT/32) */
#define NSORT   16384

__device__ __constant__ int c_lvl_off[5] = {0, 120000, 150000, 157500, 159375};
__device__ __constant__ int c_lvl_cnt[5] = {120000, 30000, 7500, 1875, 507};
__device__ __constant__ int c_lvl_k[5]   = {2000, 2000, 2000, 1875, 507};
__device__ __constant__ int c_k_off[5]   = {0, 2000, 4000, 6000, 7875};

#define AS_GLOBAL __attribute__((address_space(1)))
#define AS_LDS    __attribute__((address_space(3)))

typedef int v4i __attribute__((vector_size(16)));   // b128 payload type

// ---- gfx1250 async global->LDS copy (ASYNCcnt path) ------------------------
__device__ __forceinline__ void async_copy_b128(const void* gsrc, void* ldst) {
#if __has_builtin(__builtin_amdgcn_global_load_async_to_lds_b128)
  __builtin_amdgcn_global_load_async_to_lds_b128(
      (AS_GLOBAL v4i*)(void*)gsrc,   // global src (AS1, v4i pointee per builtin proto)
      (AS_LDS v4i*)ldst,             // LDS dst (AS3)
      0, 0);
#else
  unsigned l = (unsigned)(unsigned long long)(AS_LDS void*)ldst;
  asm volatile("global_load_async_to_lds_b128 %0, %1, off"
               :: "v"(l), "v"(gsrc) : "memory");
#endif
}
__device__ __forceinline__ void wait_async_zero() {
#if __has_builtin(__builtin_amdgcn_s_wait_asynccnt)
  __builtin_amdgcn_s_wait_asynccnt(0);
#else
  asm volatile("s_wait_asynccnt 0" ::: "memory");
#endif
}

__device__ __forceinline__ uint32_t ord32(float f) {
  uint32_t b = __float_as_uint(f);
  return (b & 0x80000000u) ? ~b : (b | 0x80000000u);
}

// ---- 1. decode + clip + valid + orderable score key ------------------------
__global__ void decode_kernel(const float* __restrict__ sc,
                              const float* __restrict__ dl,
                              const float* __restrict__ an,
                              float4* __restrict__ boxes,
                              uint32_t* __restrict__ skey,
                              uint32_t* __restrict__ val) {
  int i = blockIdx.x * blockDim.x + threadIdx.x;
  if (i >= A_TOTAL) return;
  float ax0 = an[4*i+0], ay0 = an[4*i+1], ax1 = an[4*i+2], ay1 = an[4*i+3];
  float w = ax1 - ax0, h = ay1 - ay0;
  float cx = ax0 + 0.5f*w, cy = ay0 + 0.5f*h;
  float dx = dl[4*i+0], dy = dl[4*i+1];
  float dw = fminf(dl[4*i+2], BBOX_CLIPV), dh = fminf(dl[4*i+3], BBOX_CLIPV);
  float px = cx + dx*w, py = cy + dy*h;
  float pw = expf(dw)*w, ph = expf(dh)*h;
  float x0 = fminf(fmaxf(px - 0.5f*pw, 0.f), IMG_WH);
  float y0 = fminf(fmaxf(py - 0.5f*ph, 0.f), IMG_WH);
  float x1 = fminf(fmaxf(px + 0.5f*pw, 0.f), IMG_WH);
  float y1 = fminf(fmaxf(py + 0.5f*ph, 0.f), IMG_WH);
  boxes[i] = make_float4(x0, y0, x1, y1);
  val[i]   = ((x1 - x0 >= MIN_SIZE) && (y1 - y0 >= MIN_SIZE)) ? 1u : 0u;
  skey[i]  = ord32(sc[i]);
}

// ---- 2. per-level 4-pass radix select: k-th largest key + tie quota --------
__global__ __launch_bounds__(1024) void radix_kernel(const uint32_t* __restrict__ skey,
                                                     uint32_t* __restrict__ levT,
                                                     uint32_t* __restrict__ levTie) {
  int lvl = blockIdx.x;
  int off = c_lvl_off[lvl], C = c_lvl_cnt[lvl], k = c_lvl_k[lvl];
  __shared__ uint32_t hist[256];
  __shared__ uint32_t sh_prefix, sh_need;
  if (threadIdx.x == 0) { sh_prefix = 0u; sh_need = (uint32_t)k; }
  __syncthreads();
  for (int pass = 0; pass < 4; ++pass) {
    int shift = 24 - 8*pass;
    uint32_t pmask = (pass == 0) ? 0u : (0xFFFFFFFFu << (shift + 8));
    for (int b = threadIdx.x; b < 256; b += 1024) hist[b] = 0u;
    __syncthreads();
    uint32_t prefix = sh_prefix;
    for (int j = threadIdx.x; j < C; j += 1024) {
      uint32_t u = skey[off + j];
      if ((u & pmask) == prefix) atomicAdd(&hist[(u >> shift) & 0xFFu], 1u);
    }
    __syncthreads();
    if (threadIdx.x == 0) {
      uint32_t need = sh_need, cum = 0u; int b = 255;
      for (; b >= 0; --b) { cum += hist[b]; if (cum >= need) break; }
      if (b < 0) b = 0;
      sh_need   = need - (cum - hist[b]);
      sh_prefix = prefix | ((uint32_t)b << shift);
    }
    __syncthreads();
  }
  if (threadIdx.x == 0) { levT[lvl] = sh_prefix; levTie[lvl] = sh_need; }
}

// ---- 3. deterministic scan compaction of selected indices ------------------
__global__ __launch_bounds__(1024) void compact_kernel(const uint32_t* __restrict__ skey,
                                                       const uint32_t* __restrict__ levT,
                                                       const uint32_t* __restrict__ levTie,
                                                       uint32_t* __restrict__ sel_idx) {
  int lvl = blockIdx.x;
  int off = c_lvl_off[lvl], C = c_lvl_cnt[lvl], k = c_lvl_k[lvl], ko = c_k_off[lvl];
  uint32_t T = levT[lvl], t = levTie[lvl];
  uint32_t gtot = (uint32_t)k - t;
  __shared__ uint32_t ssc[1024];
  __shared__ uint32_t base_g, base_e;
  if (threadIdx.x == 0) { base_g = 0u; base_e = 0u; }
  __syncthreads();
  for (int start = 0; start < C; start += 1024) {
    int j = start + threadIdx.x;
    bool inb = j < C;
    uint32_t u = inb ? skey[off + j] : 0u;
    bool gt = inb && (u > T);
    bool eq = inb && (u == T);
    uint32_t v = (gt ? 0x10000u : 0u) | (eq ? 1u : 0u);
    ssc[threadIdx.x] = v; __syncthreads();
    for (int s = 1; s < 1024; s <<= 1) {
      uint32_t a = (threadIdx.x >= (unsigned)s) ? ssc[threadIdx.x - s] : 0u;
      __syncthreads();
      ssc[threadIdx.x] += a;
      __syncthreads();
    }
    uint32_t incl = ssc[threadIdx.x];
    if (gt) { uint32_t n = base_g + (incl >> 16) - 1u; sel_idx[ko + n] = (uint32_t)(off + j); }
    if (eq) { uint32_t m = base_e + (incl & 0xFFFFu) - 1u;
              if (m < t) sel_idx[ko + gtot + m] = (uint32_t)(off + j); }
    __syncthreads();
    if (threadIdx.x == 0) { uint32_t tot = ssc[1023]; base_g += tot >> 16; base_e += tot & 0xFFFFu; }
    __syncthreads();
  }
}

// ---- 4. single-WGP bitonic sort of 16384 u64 keys in 128KB LDS -------------
__global__ __launch_bounds__(1024) void sort_kernel(const float* __restrict__ sc,
                                                    const uint32_t* __restrict__ sel_idx,
                                                    const uint32_t* __restrict__ val,
                                                    const float4* __restrict__ boxes,
                                                    float* __restrict__ s_srt,
                                                    float4* __restrict__ b_srt) {
  __shared__ unsigned long long sk[NSORT];   // 128 KB: fits CDNA5's 320KB LDS
  for (int p = threadIdx.x; p < NSORT; p += 1024) {
    unsigned long long key = 0ull;
    if (p < KTOT) {
      uint32_t a = sel_idx[p];
      float s = val[a] ? sc[a] : NEG_INF;
      key = ((unsigned long long)ord32(s) << 32) |
            (unsigned long long)(0xFFFFFFFFu - (uint32_t)p);   // tie: lower pos first
    }
    sk[p] = key;
  }
  __syncthreads();
  for (int size = 2; size <= NSORT; size <<= 1) {
    for (int stride = size >> 1; stride > 0; stride >>= 1) {
      for (int i = threadIdx.x; i < NSORT; i += 1024) {
        int ixj = i ^ stride;
        if (ixj > i) {
          bool up = ((i & size) == 0);   // descending overall
          unsigned long long a = sk[i], b = sk[ixj];
          if ((a < b) == up) { sk[i] = b; sk[ixj] = a; }
        }
      }
      __syncthreads();
    }
  }
  for (int r = threadIdx.x; r < KPAD; r += 1024) {
    if (r < KTOT) {
      uint32_t pos = 0xFFFFFFFFu - (uint32_t)(sk[r] & 0xFFFFFFFFull);
      uint32_t a = sel_idx[pos];
      s_srt[r] = val[a] ? sc[a] : NEG_INF;
      b_srt[r] = boxes[a];
    } else {
      s_srt[r] = NEG_INF;
      b_srt[r] = make_float4(0.f, 0.f, 0.f, 0.f);   // inert pad boxes
    }
  }
}

// ---- 5. IoU suppression mask: 8 waves share one async-staged col tile ------
// Block = 256 threads (8 wave32). Wave w handles row tile yb = blockIdx.y*8+w
// against col tile xb = blockIdx.x. Column boxes staged once per block via
// gfx1250 async global->LDS; each lane emits a 32-bit suppression word.
__global__ __launch_bounds__(256) void mask_kernel(const float4* __restrict__ b_srt,
                                                   uint32_t* __restrict__ mask) {
  int xb  = blockIdx.x;
  int yb0 = blockIdx.y * 8;
  if (xb < yb0) return;                      // whole block below diagonal
  __shared__ float4 cb[32];
  __shared__ float  carea[32];
  int tid  = threadIdx.x;
  int lane = tid & 31;
  int wv   = tid >> 5;
  if (tid < 32) {                            // wave 0 stages the column tile
    async_copy_b128(&b_srt[xb*32 + tid], &cb[tid]);
    wait_async_zero();
  }
  __syncthreads();
  if (tid < 32) {
    float4 c = cb[tid];
    carea[tid] = (c.z - c.x) * (c.w - c.y);
  }
  __syncthreads();
  int yb = yb0 + wv;
  if (yb > xb || yb >= NW) return;           // per-wave skip (after barriers)
  int r = yb*32 + lane;                      // r < KPAD always
  float4 rb = b_srt[r];
  float ra = (rb.z - rb.x) * (rb.w - rb.y);
  uint32_t wrd = 0u;
  #pragma unroll 8
  for (int j = 0; j < 32; ++j) {
    int cidx = xb*32 + j;
    float4 q = cb[j];
    float ix0 = fmaxf(rb.x, q.x), iy0 = fmaxf(rb.y, q.y);
    float ix1 = fminf(rb.z, q.z), iy1 = fminf(rb.w, q.w);
    float iw = fmaxf(ix1 - ix0, 0.f), ih = fmaxf(iy1 - iy0, 0.f);
    float inter = iw * ih;
    float iou = inter / (ra + carea[j] - inter);  // NaN compares false
    if ((cidx > r) && (iou > NMS_TH)) wrd |= (1u << j);
  }
  if (r < KTOT) mask[(size_t)r * NW + xb] = wrd;
}

// ---- 6. serial NMS reduction on a single wave32 ----------------------------
// The inherently serial chain. Per-iteration barrier forces the compiler to
// re-read LDS words OR-written by sibling lanes (single-wave workgroup: the
// barrier degenerates to a no-op + dscnt wait). Next row is prefetched so the
// HBM fetch overlaps the current iteration's check/OR.
__global__ __launch_bounds__(32) void nms_reduce_kernel(const uint32_t* __restrict__ mask,
                                                        const float* __restrict__ s_srt,
                                                        uint32_t* __restrict__ keep) {
  __shared__ uint32_t remv[NW];
  int t = threadIdx.x;
  for (int w = t; w < NW; w += 32) remv[w] = 0u;
  __syncthreads();
  for (int i = 0; i < KTOT; ++i) {
    if (i + 1 < KTOT) {                       // prefetch next row (~9 lanes cover it)
      const uint32_t* nrow = mask + (size_t)(i + 1) * NW;
      if (t * 32 < NW) __builtin_prefetch(&nrow[t * 32], 0, 0);
    }
    uint32_t w_i = (uint32_t)i >> 5, b_i = (uint32_t)i & 31u;
    uint32_t word = remv[w_i];                // uniform read
    bool alive = (((word >> b_i) & 1u) == 0u) && (s_srt[i] > NEG_INF);
    if (alive) {
      const uint32_t* row = mask + (size_t)i * NW;
      for (uint32_t w = w_i + (uint32_t)t; w < NW; w += 32) remv[w] |= row[w];
    }
    if (t == 0) keep[i] = alive ? 1u : 0u;
    __syncthreads();                          // LDS visibility for next iteration
  }
}

// ---- 7. stable compaction of kept boxes -> top-1000 output -----------------
__global__ __launch_bounds__(1024) void output_kernel(const uint32_t* __restrict__ keep,
                                                      const float* __restrict__ s_srt,
                                                      const float4* __restrict__ b_srt,
                                                      float* __restrict__ out) {
  __shared__ uint32_t ssc[1024];
  __shared__ uint32_t sm, base_k;
  if (threadIdx.x == 0) { sm = 0u; base_k = 0u; }
  __syncthreads();
  uint32_t loc = 0u;
  for (int j = threadIdx.x; j < KTOT; j += 1024) loc += keep[j];
  atomicAdd(&sm, loc);
  __syncthreads();
  uint32_t m = sm;                 // total kept
  __syncthreads();
  float* out_b = out;              // [1000,4]
  float* out_s = out + 4000;       // [1000]
  for (int start = 0; start < KTOT; start += 1024) {
    int j = start + threadIdx.x;
    bool inb = j < KTOT;
    uint32_t kp = inb ? keep[j] : 0u;
    ssc[threadIdx.x] = kp; __syncthreads();
    for (int s = 1; s < 1024; s <<= 1) {
      uint32_t a = (threadIdx.x >= (unsigned)s) ? ssc[threadIdx.x - s] : 0u;
      __syncthreads();
      ssc[threadIdx.x] += a;
      __syncthreads();
    }
    uint32_t incl = ssc[threadIdx.x];
    if (inb) {
      uint32_t krank  = base_k + incl - kp;          // exclusive kept rank
      uint32_t nkrank = (uint32_t)j - krank;         // exclusive not-kept rank
      if (kp) {
        if (krank < 1000u) {
          float4 b = b_srt[j];
          out_b[4*krank+0] = b.x; out_b[4*krank+1] = b.y;
          out_b[4*krank+2] = b.z; out_b[4*krank+3] = b.w;
          out_s[krank] = s_srt[j];
        }
      } else {
        uint32_t slot = m + nkrank;                  // -inf padding, lowest idx first
        if (slot < 1000u) {
          float4 b = b_srt[j];
          out_b[4*slot+0] = b.x; out_b[4*slot+1] = b.y;
          out_b[4*slot+2] = b.z; out_b[4*slot+3] = b.w;
          out_s[slot] = NEG_INF;
        }
      }
    }
    __syncthreads();
    if (threadIdx.x == 0) base_k += ssc[1023];
    __syncthreads();
  }
}

extern "C" void kernel_launch(void* const* d_in, const int* in_sizes, int n_in,
                              void* d_out, int out_size, void* d_ws, size_t ws_size,
                              hipStream_t stream) {
  (void)in_sizes; (void)n_in; (void)out_size; (void)ws_size;
  const float* sc = (const float*)d_in[0];   // scores  (1, A_TOTAL)
  const float* dl = (const float*)d_in[1];   // deltas  (A_TOTAL, 4)
  const float* an = (const float*)d_in[2];   // anchors (A_TOTAL, 4)

  char* ws = (char*)d_ws;                    // ~12.3 MB used, 8B-aligned offsets
  float4*   boxes   = (float4*)  (ws + 0);         // A_TOTAL*16 = 2558112
  uint32_t* skey    = (uint32_t*)(ws + 2558112);   // A_TOTAL*4
  uint32_t* val     = (uint32_t*)(ws + 3197640);   // A_TOTAL*4
  uint32_t* levT    = (uint32_t*)(ws + 3837168);   // 8*4
  uint32_t* levTie  = (uint32_t*)(ws + 3837200);   // 8*4
  uint32_t* sel_idx = (uint32_t*)(ws + 3837232);   // KPAD*4
  float*    s_srt   = (float*)   (ws + 3870768);   // KPAD*4
  float4*   b_srt   = (float4*)  (ws + 3904304);   // KPAD*16
  uint32_t* keep    = (uint32_t*)(ws + 4038448);   // KPAD*4
  uint32_t* mask    = (uint32_t*)(ws + 4071984);   // KPAD*NW*4 = 8786432

  decode_kernel<<<(A_TOTAL + 255)/256, 256, 0, stream>>>(sc, dl, an, boxes, skey, val);
  radix_kernel<<<5, 1024, 0, stream>>>(skey, levT, levTie);
  compact_kernel<<<5, 1024, 0, stream>>>(skey, levT, levTie, sel_idx);
  sort_kernel<<<1, 1024, 0, stream>>>(sc, sel_idx, val, boxes, s_srt, b_srt);
  mask_kernel<<<dim3(NW, (NW + 7) / 8), 256, 0, stream>>>(b_srt, mask);
  nms_reduce_kernel<<<1, 32, 0, stream>>>(mask, s_srt, keep);
  output_kernel<<<1, 1024, 0, stream>>>(keep, s_srt, b_srt, (float*)d_out);
}